// MultiboxLoss_44659069944172
// MI455X (gfx1250) — compile-verified
//
#include <hip/hip_runtime.h>
#include <math.h>

#define B_ 128
#define G_ 16
#define A_ 8732
#define C_ 21
#define THR_ 0.5f
#define NEG_POS_ 3
#define VAR0_ 0.1f
#define VAR1_ 0.2f
#define NBLK_ ((A_ + 255) / 256)

typedef float v2f __attribute__((ext_vector_type(2)));
typedef float v8f __attribute__((ext_vector_type(8)));
typedef unsigned int v4u __attribute__((ext_vector_type(4)));
typedef int v8i __attribute__((ext_vector_type(8)));
typedef int v4i __attribute__((ext_vector_type(4)));

// ---------------------------------------------------------------------------
// Block-wide (256-thread) f32 sum on the CDNA5 matrix pipe:
// V_WMMA_F32_16X16X4_F32 with all-ones B gives 16 row sums of a 16x4 f32
// A-tile; 256 values = 4 tiles accumulated into one C. Full f32 precision.
// WMMA requires EXEC all-1s: only wave 0 (all 32 lanes) issues it.
// ---------------------------------------------------------------------------
__device__ __forceinline__ float block_reduce256_wmma(float v, float* lds) {
  const int tid = threadIdx.x;
  lds[tid] = v;
  __syncthreads();
  if (tid < 32) {
    const int m  = tid & 15;
    const int kh = tid >> 4;
    v8f acc = {};
    v2f ones; ones[0] = 1.0f; ones[1] = 1.0f;
#pragma unroll
    for (int t = 0; t < 4; ++t) {
      v2f a;
      a[0] = lds[t * 64 + m * 4 + kh * 2 + 0];
      a[1] = lds[t * 64 + m * 4 + kh * 2 + 1];
      acc = __builtin_amdgcn_wmma_f32_16x16x4_f32(false, a, false, ones,
                                                  (short)0, acc, false, false);
    }
    float s = acc[0] + acc[1] + acc[2] + acc[3] +
              acc[4] + acc[5] + acc[6] + acc[7];
    float tot = __shfl(s, 0, 32) + __shfl(s, 16, 32);
    if (tid == 0) lds[0] = tot;
  }
  __syncthreads();
  float r = lds[0];
  __syncthreads();
  return r;
}

// ---------------------------------------------------------------------------
// K1: per-batch IoU match. One block per b, 256 threads stream all anchors.
// ---------------------------------------------------------------------------
__global__ void k_match(const float* __restrict__ tboxes,   // (B,G,4) point form
                        const float* __restrict__ anchors,  // (A,4) cx,cy,w,h
                        float* __restrict__ gt_overlap,
                        int* __restrict__ gt_index,
                        int* __restrict__ anchor_index) {
  __shared__ float sb[G_ * 4];
  __shared__ float s_gmax[G_ * 256];
  __shared__ int   s_gidx[G_ * 256];
  const int b = blockIdx.x, tid = threadIdx.x;
  if (tid < G_ * 4) sb[tid] = tboxes[b * G_ * 4 + tid];
  __syncthreads();

  float gmax[G_]; int gidx[G_];
#pragma unroll
  for (int g = 0; g < G_; ++g) { gmax[g] = -1.0f; gidx[g] = 0; }

  for (int a = tid; a < A_; a += 256) {
    const float cx = anchors[a*4+0], cy = anchors[a*4+1];
    const float w  = anchors[a*4+2], h  = anchors[a*4+3];
    const float ax1 = cx - 0.5f*w, ay1 = cy - 0.5f*h;
    const float ax2 = cx + 0.5f*w, ay2 = cy + 0.5f*h;
    const float areaA = w * h;
    float best = -1.0f; int bestg = 0;
#pragma unroll
    for (int g = 0; g < G_; ++g) {
      const float bx1 = sb[g*4+0], by1 = sb[g*4+1];
      const float bx2 = sb[g*4+2], by2 = sb[g*4+3];
      float iw = fmaxf(fminf(ax2, bx2) - fmaxf(ax1, bx1), 0.0f);
      float ih = fmaxf(fminf(ay2, by2) - fmaxf(ay1, by1), 0.0f);
      const float inter = iw * ih;
      const float areaB = (bx2 - bx1) * (by2 - by1);
      const float iou = inter / (areaA + areaB - inter);
      if (iou > best)    { best = iou; bestg = g; }
      if (iou > gmax[g]) { gmax[g] = iou; gidx[g] = a; }
    }
    gt_overlap[(size_t)b*A_ + a] = best;
    gt_index  [(size_t)b*A_ + a] = bestg;
  }
#pragma unroll
  for (int g = 0; g < G_; ++g) { s_gmax[g*256+tid] = gmax[g]; s_gidx[g*256+tid] = gidx[g]; }
  __syncthreads();
  if (tid < G_) {
    float bm = -1.0f; int bi = 0x7fffffff;
    for (int t = 0; t < 256; ++t) {
      const float v = s_gmax[tid*256+t]; const int i2 = s_gidx[tid*256+t];
      if (v > bm || (v == bm && i2 < bi)) { bm = v; bi = i2; }
    }
    anchor_index[b * G_ + tid] = bi;
  }
}

// K2: force-match overrides (gt_index loop sequential: last GT wins)
__global__ void k_override(const int* __restrict__ anchor_index,
                           float* __restrict__ gt_overlap,
                           int* __restrict__ gt_index) {
  const int b = blockIdx.x, tid = threadIdx.x;
  if (tid < G_) gt_overlap[(size_t)b*A_ + anchor_index[b*G_ + tid]] = 1.0f;
  if (tid == 0)
    for (int i = 0; i < G_; ++i)
      gt_index[(size_t)b*A_ + anchor_index[b*G_ + i]] = i;
}

// ---------------------------------------------------------------------------
// K3: stage the block's 256x21 confidence tile into LDS via the gfx1250
// async-to-LDS path (coalesced b32 copies, ASYNCcnt), then smooth-L1 on
// positives + log-softmax NLL from LDS. Per-block partials reduced via WMMA.
// ---------------------------------------------------------------------------
__global__ void k_loss(const float* __restrict__ pred_loc,
                       const float* __restrict__ pred_conf,
                       const float* __restrict__ tboxes,
                       const int*   __restrict__ tlabels,
                       const float* __restrict__ anchors,
                       const float* __restrict__ gt_overlap,
                       const int*   __restrict__ gt_index,
                       float* __restrict__ aux,
                       float* __restrict__ loc_part,
                       float* __restrict__ pnll_part,
                       float* __restrict__ pcnt_part) {
  __shared__ float s_conf[256 * C_];   // 21 KB tile
  __shared__ float lds[256];
  const int b = blockIdx.y;
  const int a = blockIdx.x * 256 + threadIdx.x;
  const bool valid = (a < A_);
  const int aa = valid ? a : 0;

  // --- async stage: 5376 dwords, coalesced, clamped at buffer end ---------
  {
    const size_t tile_dw = ((size_t)b * A_ + (size_t)blockIdx.x * 256) * C_;
    const size_t max_dw  = (size_t)B_ * A_ * C_ - 1;
    const unsigned lbase = (unsigned)(uintptr_t)s_conf;   // wave-relative LDS byte addr
#pragma unroll
    for (int j = 0; j < C_; ++j) {            // 256*21 dwords, thread d = tid + j*256
      const int d = threadIdx.x + j * 256;
      size_t gdw = tile_dw + (size_t)d;
      if (gdw > max_dw) gdw = max_dw;
      const unsigned voff  = (unsigned)(gdw * 4u);
      const unsigned laddr = lbase + (unsigned)d * 4u;
      asm volatile("global_load_async_to_lds_b32 %0, %1, %2"
                   :: "v"(laddr), "v"(voff), "s"(pred_conf) : "memory");
    }
    asm volatile("s_wait_asynccnt 0x0" ::: "memory");
  }
  __builtin_prefetch(pred_conf + ((size_t)b * A_ + (size_t)blockIdx.x * 256) * C_
                               + 256 * C_, 0, 0);  // warm L2 for next tile
  __syncthreads();

  const float ov = gt_overlap[(size_t)b*A_ + aa];
  const int   gi = gt_index  [(size_t)b*A_ + aa];
  const int  lab = tlabels[b*G_ + gi];
  const int conf = (ov < THR_) ? 0 : (lab + 1);
  const bool is_pos = (conf > 0);

  const float* cp = &s_conf[threadIdx.x * C_];  // stride 21 dwords: bank-conflict-free
  float mx = cp[0];
#pragma unroll
  for (int c = 1; c < C_; ++c) mx = fmaxf(mx, cp[c]);
  float se = 0.0f;
#pragma unroll
  for (int c = 0; c < C_; ++c) se += expf(cp[c] - mx);
  const float lse = logf(se) + mx;
  const float nll = lse - cp[conf];

  float lloc = 0.0f, pnll = 0.0f, pcnt = 0.0f;
  if (valid && is_pos) {
    const float bx1 = tboxes[(b*G_+gi)*4+0], by1 = tboxes[(b*G_+gi)*4+1];
    const float bx2 = tboxes[(b*G_+gi)*4+2], by2 = tboxes[(b*G_+gi)*4+3];
    const float acx = anchors[aa*4+0], acy = anchors[aa*4+1];
    const float aw  = anchors[aa*4+2], ah  = anchors[aa*4+3];
    float lt[4];
    lt[0] = (0.5f*(bx1+bx2) - acx) / (aw * VAR0_);
    lt[1] = (0.5f*(by1+by2) - acy) / (ah * VAR0_);
    lt[2] = logf((bx2 - bx1) / aw) / VAR1_;
    lt[3] = logf((by2 - by1) / ah) / VAR1_;
#pragma unroll
    for (int k = 0; k < 4; ++k) {
      const float d  = pred_loc[((size_t)b*A_ + aa)*4 + k] - lt[k];
      const float ad = fabsf(d);
      lloc += (ad < 1.0f) ? 0.5f*d*d : ad - 0.5f;
    }
    pnll = nll; pcnt = 1.0f;
  }
  if (valid) aux[(size_t)b*A_ + a] = is_pos ? 0.0f : nll;

  const float s1 = block_reduce256_wmma(lloc, lds);
  const float s2 = block_reduce256_wmma(pnll, lds);
  const float s3 = block_reduce256_wmma(pcnt, lds);
  if (threadIdx.x == 0) {
    loc_part [b*NBLK_ + blockIdx.x] = s1;
    pnll_part[b*NBLK_ + blockIdx.x] = s2;
    pcnt_part[b*NBLK_ + blockIdx.x] = s3;
  }
}

// K4: fold per-block partials into per-batch accumulators (deterministic)
__global__ void k_reduceB(const float* __restrict__ loc_part,
                          const float* __restrict__ pnll_part,
                          const float* __restrict__ pcnt_part,
                          float* __restrict__ loss_loc_acc,
                          float* __restrict__ pos_nll_acc,
                          float* __restrict__ num_pos_acc) {
  const int b = threadIdx.x;
  float l = 0.0f, p = 0.0f, c = 0.0f;
  for (int i = 0; i < NBLK_; ++i) {
    l += loc_part [b*NBLK_ + i];
    p += pnll_part[b*NBLK_ + i];
    c += pcnt_part[b*NBLK_ + i];
  }
  loss_loc_acc[b] = l; pos_nll_acc[b] = p; num_pos_acc[b] = c;
}

// ---------------------------------------------------------------------------
// K5: hard-negative mining = top-k sum of aux (aux >= 0).
// The 35 KB aux row is staged into LDS ONCE via the Tensor Data Mover
// (1 x 8732 f32 tile descriptor, TENSORcnt), then the 31-step bitwise
// binary search for the k-th largest value scans LDS only.
// sum = sum(aux > t) + (k - count(aux > t)) * t   handles ties exactly.
// ---------------------------------------------------------------------------
__global__ void k_topk(const float* __restrict__ aux,
                       const float* __restrict__ num_pos_acc,
                       float* __restrict__ neg_sum) {
  __shared__ float s_aux[A_];          // 34,928 B
  __shared__ float lds[256];
  const int b = blockIdx.x, tid = threadIdx.x;

  // --- TDM stage: one descriptor, issued by wave 0 only --------------------
  if (tid < 32) {
    const unsigned lbase = (unsigned)(uintptr_t)s_aux;
    const unsigned long long ga =
        (unsigned long long)(uintptr_t)(aux + (size_t)b * A_);
    v4u g0;
    g0[0] = 1u;                                         // count=1 (valid D#)
    g0[1] = lbase;                                      // lds_addr
    g0[2] = (unsigned)(ga & 0xffffffffu);               // global_addr[31:0]
    g0[3] = (unsigned)((ga >> 32) & 0x01ffffffu)        // global_addr[56:32]
            | (2u << 30);                               // type = 2 ("image")
    v8i g1;
    g1[0] = (int)(2u << 16);                            // data_size=4B, mask=0
    g1[1] = (int)(((unsigned)A_ & 0xffffu) << 16);      // tensor_dim0[15:0]
    g1[2] = (int)((((unsigned)A_ >> 16) & 0xffffu)      // tensor_dim0[31:16]
                  | (1u << 16));                        // tensor_dim1 = 1
    g1[3] = (int)((unsigned)A_ << 16);                  // tile_dim0 = A
    g1[4] = 1;                                          // tile_dim1=1, tile_dim2=0
    g1[5] = A_;                                         // tensor_dim0_stride
    g1[6] = 0;
    g1[7] = 0;
    v4i gz4 = {0, 0, 0, 0};
    v8i gz8 = {0, 0, 0, 0, 0, 0, 0, 0};
    // amdgpu-toolchain (clang-23) arity: 6 args, extra int32x8 group + cpol
    __builtin_amdgcn_tensor_load_to_lds(g0, g1, gz4, gz4, gz8, 0);
    __builtin_amdgcn_s_wait_tensorcnt((short)0);
  }
  __syncthreads();

  const int np = (int)(num_pos_acc[b] + 0.5f);
  int k = NEG_POS_ * np; if (k > A_ - 1) k = A_ - 1;
  if (k <= 0) { if (tid == 0) neg_sum[b] = 0.0f; return; }  // uniform exit

  unsigned u = 0;
  for (int bit = 30; bit >= 0; --bit) {
    const unsigned cand = u | (1u << bit);
    const float t = __uint_as_float(cand);
    float cnt = 0.0f;
    for (int a = tid; a < A_; a += 256) cnt += (s_aux[a] >= t) ? 1.0f : 0.0f;
    const float total = block_reduce256_wmma(cnt, lds);   // exact: A < 2^24
    if (total >= (float)k) u = cand;
  }
  const float t = __uint_as_float(u);
  float sgt = 0.0f, cgt = 0.0f;
  for (int a = tid; a < A_; a += 256) {
    const float v = s_aux[a];
    if (v > t) { sgt += v; cgt += 1.0f; }
  }
  const float Sgt = block_reduce256_wmma(sgt, lds);
  const float Cgt = block_reduce256_wmma(cgt, lds);
  if (tid == 0) neg_sum[b] = Sgt + ((float)k - Cgt) * t;
}

// K6: final scalars
__global__ void k_final(const float* __restrict__ loss_loc_acc,
                        const float* __restrict__ pos_nll_acc,
                        const float* __restrict__ neg_sum,
                        const float* __restrict__ num_pos_acc,
                        float* __restrict__ out) {
  __shared__ float lds[256];
  const int tid = threadIdx.x;
  float l = 0.0f, c = 0.0f, n = 0.0f;
  if (tid < B_) {
    l = loss_loc_acc[tid];
    c = pos_nll_acc[tid] + neg_sum[tid];
    n = num_pos_acc[tid];
  }
  const float L = block_reduce256_wmma(l, lds);
  const float Cs = block_reduce256_wmma(c, lds);
  const float N = block_reduce256_wmma(n, lds);
  if (tid == 0) { out[0] = L / N; out[1] = Cs / N; }
}

extern "C" void kernel_launch(void* const* d_in, const int* in_sizes, int n_in,
                              void* d_out, int out_size, void* d_ws, size_t ws_size,
                              hipStream_t stream) {
  (void)in_sizes; (void)n_in; (void)out_size; (void)ws_size;
  const float* pred_loc  = (const float*)d_in[0];  // (B,A,4)
  const float* pred_conf = (const float*)d_in[1];  // (B,A,C)
  const float* tboxes    = (const float*)d_in[2];  // (B,G,4)
  const int*   tlabels   = (const int*)d_in[3];    // (B,G)
  const float* anchors   = (const float*)d_in[4];  // (A,4)
  float* out = (float*)d_out;

  char* w = (char*)d_ws;
  float* gt_overlap   = (float*)w; w += sizeof(float) * (size_t)B_ * A_;
  int*   gt_index     = (int*)w;   w += sizeof(int)   * (size_t)B_ * A_;
  float* aux          = (float*)w; w += sizeof(float) * (size_t)B_ * A_;
  int*   anchor_index = (int*)w;   w += sizeof(int)   * B_ * G_;
  float* loc_part     = (float*)w; w += sizeof(float) * B_ * NBLK_;
  float* pnll_part    = (float*)w; w += sizeof(float) * B_ * NBLK_;
  float* pcnt_part    = (float*)w; w += sizeof(float) * B_ * NBLK_;
  float* loss_loc_acc = (float*)w; w += sizeof(float) * B_;
  float* pos_nll_acc  = (float*)w; w += sizeof(float) * B_;
  float* num_pos_acc  = (float*)w; w += sizeof(float) * B_;
  float* neg_sum      = (float*)w; w += sizeof(float) * B_;

  k_match   <<<B_, 256, 0, stream>>>(tboxes, anchors, gt_overlap, gt_index, anchor_index);
  k_override<<<B_,  32, 0, stream>>>(anchor_index, gt_overlap, gt_index);
  dim3 g3(NBLK_, B_);
  k_loss    <<<g3, 256, 0, stream>>>(pred_loc, pred_conf, tboxes, tlabels, anchors,
                                     gt_overlap, gt_index, aux,
                                     loc_part, pnll_part, pcnt_part);
  k_reduceB <<<1,  B_,  0, stream>>>(loc_part, pnll_part, pcnt_part,
                                     loss_loc_acc, pos_nll_acc, num_pos_acc);
  k_topk    <<<B_, 256, 0, stream>>>(aux, num_pos_acc, neg_sum);
  k_final   <<<1,  256, 0, stream>>>(loss_loc_acc, pos_nll_acc, neg_sum, num_pos_acc, out);
}